// RecSysGNN_57578331570196
// MI455X (gfx1250) — compile-verified
//
#include <hip/hip_runtime.h>
#include <hip/hip_bf16.h>
#include <stdint.h>

// ---------------------------------------------------------------------------
// LightGCN propagation for MI455X (gfx1250).
// Sparse gather/scatter core: wave-per-edge, lane-per-float2 => coalesced 256B
// row gathers + global_atomic_add_f32 scatter; edge metadata staged into LDS
// via gfx1250 async-to-LDS (ASYNCcnt), source rows prefetched with
// global_prefetch. Final layer-mean pass runs on the matrix pipe:
// 4x chained v_wmma_f32_16x16x4_f32 per 16x16 tile computing (acc+x3)*0.25
// exactly (one 0.25*S product per element, rest zeros).
// ---------------------------------------------------------------------------

#if defined(__has_builtin)
#if __has_builtin(__builtin_amdgcn_global_load_async_to_lds_b128) && \
    __has_builtin(__builtin_amdgcn_s_wait_asynccnt)
#define USE_ASYNC_LDS 1
#endif
#endif
#ifndef USE_ASYNC_LDS
#define USE_ASYNC_LDS 0
#endif

#if USE_ASYNC_LDS
typedef int v4i_t __attribute__((vector_size(16)));  // param type per hipcc diag
#endif

typedef float v2f __attribute__((ext_vector_type(2)));
typedef float v8f __attribute__((ext_vector_type(8)));

#define PROP_TILE 1024  // edges of metadata staged per block-batch (12 KB LDS)

__global__ __launch_bounds__(256) void k_zero(float* __restrict__ p, int n) {
  int i = blockIdx.x * blockDim.x + threadIdx.x;
  int s = gridDim.x * blockDim.x;
  int n4 = n >> 2;
  float4* p4 = (float4*)p;
  for (int k = i; k < n4; k += s) p4[k] = make_float4(0.f, 0.f, 0.f, 0.f);
  for (int k = (n4 << 2) + i; k < n; k += s) p[k] = 0.f;
}

__global__ __launch_bounds__(256) void k_degree(const int* __restrict__ to,
                                                float* __restrict__ deg, int E) {
  int i = blockIdx.x * blockDim.x + threadIdx.x;
  int s = gridDim.x * blockDim.x;
  for (; i < E; i += s) unsafeAtomicAdd(&deg[to[i]], 1.0f);
}

__global__ __launch_bounds__(256) void k_rsqrt_inplace(float* __restrict__ d, int n) {
  int i = blockIdx.x * blockDim.x + threadIdx.x;
  int s = gridDim.x * blockDim.x;
  for (; i < n; i += s) {
    float v = d[i];
    d[i] = (v > 0.f) ? (1.0f / sqrtf(v)) : 0.f;
  }
}

__global__ __launch_bounds__(256) void k_norm(const int* __restrict__ frm,
                                              const int* __restrict__ to,
                                              const float* __restrict__ dinv,
                                              float* __restrict__ nrm, int E) {
  int i = blockIdx.x * blockDim.x + threadIdx.x;
  int s = gridDim.x * blockDim.x;
  for (; i < E; i += s) nrm[i] = dinv[frm[i]] * dinv[to[i]];
}

// out[0:n64) = emb (first tuple element), out[n64:2*n64) = emb (acc init).
__global__ __launch_bounds__(256) void k_init_out(const float4* __restrict__ emb4,
                                                  float4* __restrict__ out4, int n644) {
  int i = blockIdx.x * blockDim.x + threadIdx.x;
  int s = gridDim.x * blockDim.x;
  for (; i < n644; i += s) {
    float4 v = emb4[i];
    out4[i] = v;
    out4[n644 + i] = v;
  }
}

// acc = (acc + x) * scale
__global__ __launch_bounds__(256) void k_accum_scale(float4* __restrict__ acc,
                                                     const float4* __restrict__ x,
                                                     int n4, float sc) {
  int i = blockIdx.x * blockDim.x + threadIdx.x;
  int s = gridDim.x * blockDim.x;
  for (; i < n4; i += s) {
    float4 a = acc[i], v = x[i];
    a.x = (a.x + v.x) * sc;
    a.y = (a.y + v.y) * sc;
    a.z = (a.z + v.z) * sc;
    a.w = (a.w + v.w) * sc;
    acc[i] = a;
  }
}

// One wave (32 lanes) per edge; lane handles 2 consecutive floats of the
// 64-wide row. Edge metadata staged per-tile into LDS (async path on gfx1250).
__global__ __launch_bounds__(256) void k_prop(const float* __restrict__ x,
                                              float* __restrict__ xn,
                                              const int* __restrict__ frm,
                                              const int* __restrict__ to,
                                              const float* __restrict__ nrm, int E) {
  __shared__ alignas(16) int s_frm[PROP_TILE];
  __shared__ alignas(16) int s_to[PROP_TILE];
  __shared__ alignas(16) float s_nrm[PROP_TILE];

  const int lane = threadIdx.x & 31;            // wave32
  const int wave = threadIdx.x >> 5;            // 0..7
  const int wavesPerBlk = blockDim.x >> 5;

  for (int base = blockIdx.x * PROP_TILE; base < E; base += gridDim.x * PROP_TILE) {
    int tileN = E - base;
    if (tileN > PROP_TILE) tileN = PROP_TILE;

#if USE_ASYNC_LDS
    {
      const int chunks = tileN >> 2;  // 16B chunks
      for (int c = threadIdx.x; c < chunks; c += blockDim.x) {
        const int o = c << 2;
        __builtin_amdgcn_global_load_async_to_lds_b128(
            (v4i_t*)(frm + base + o), (v4i_t*)(s_frm + o), 0, 0);
        __builtin_amdgcn_global_load_async_to_lds_b128(
            (v4i_t*)(to + base + o), (v4i_t*)(s_to + o), 0, 0);
        __builtin_amdgcn_global_load_async_to_lds_b128(
            (v4i_t*)(nrm + base + o), (v4i_t*)(s_nrm + o), 0, 0);
      }
      for (int r = (chunks << 2) + (int)threadIdx.x; r < tileN; r += blockDim.x) {
        s_frm[r] = frm[base + r];
        s_to[r]  = to[base + r];
        s_nrm[r] = nrm[base + r];
      }
      __builtin_amdgcn_s_wait_asynccnt(0);
    }
#else
    for (int r = threadIdx.x; r < tileN; r += blockDim.x) {
      s_frm[r] = frm[base + r];
      s_to[r]  = to[base + r];
      s_nrm[r] = nrm[base + r];
    }
#endif
    __syncthreads();

    for (int j = wave; j < tileN; j += wavesPerBlk) {
      // Prefetch next edge's source row into WGP cache (global_prefetch_b8).
      int jn = j + wavesPerBlk;
      if (jn < tileN) {
        const char* pf = (const char*)(x + (size_t)s_frm[jn] * 64);
        __builtin_prefetch(pf + ((lane & 1) << 7), 0, 3);
      }

      int f = s_frm[j];     // LDS same-address broadcast, conflict-free
      int t = s_to[j];
      float w = s_nrm[j];

      const float2* src = (const float2*)(x + (size_t)f * 64);
      float2 v = src[lane];                     // 256B coalesced per wave
      float* dst = xn + (size_t)t * 64 + (lane << 1);
      unsafeAtomicAdd(dst, w * v.x);            // global_atomic_add_f32 (no rtn)
      unsafeAtomicAdd(dst + 1, w * v.y);
    }
    __syncthreads();
  }
}

// Final pass on the matrix pipe: acc = (acc + xlast) * 0.25 per 16x16 tile.
// D = sum_j A_j x B_j (+C chained), A_j = S[:,4j:4j+4], B_j = 0.25*selector.
// Exact: each D element = one 0.25*S product + zeros.
__global__ __launch_bounds__(256) void k_wmma_finish(const float* __restrict__ xlast,
                                                     float* __restrict__ acc,
                                                     int rowTiles) {
  const int lane = threadIdx.x & 31;
  const int wid = (blockIdx.x * blockDim.x + threadIdx.x) >> 5;  // global wave id
  const int nw = (gridDim.x * blockDim.x) >> 5;
  const int nTiles = rowTiles * 4;  // 64 cols = 4 col-tiles of 16

  const int half = lane >> 4;  // 0: K=0,1 / rows r..r+7 ; 1: K=2,3 / rows r+8..
  const int l15 = lane & 15;

  for (int tile = wid; tile < nTiles; tile += nw) {  // wave-uniform => EXEC all 1s
    const int rt = tile >> 2;
    const int c0 = (tile & 3) << 4;
    const size_t rowoff = (size_t)(rt * 16 + l15) * 64 + c0;  // A row per lane

    v8f d = {};  // C starts at zero
#pragma unroll
    for (int j = 0; j < 4; ++j) {
      const int col = 4 * j + 2 * half;
      float2 a1 = *(const float2*)(acc + rowoff + col);
      float2 a2 = *(const float2*)(xlast + rowoff + col);
      v2f A;
      A.x = a1.x + a2.x;
      A.y = a1.y + a2.y;
      v2f B;
      B.x = (l15 == col) ? 0.25f : 0.0f;      // B[k=2*half  ][n=l15]
      B.y = (l15 == col + 1) ? 0.25f : 0.0f;  // B[k=2*half+1][n=l15]
      d = __builtin_amdgcn_wmma_f32_16x16x4_f32(false, A, false, B, (short)0, d,
                                                false, false);
    }
    // D: VGPR r, lane: row = rt*16 + r + half*8, col = c0 + l15
    const size_t base = (size_t)(rt * 16 + half * 8) * 64 + c0 + l15;
#pragma unroll
    for (int r = 0; r < 8; ++r) acc[base + (size_t)r * 64] = d[r];
  }
}

static inline int ngrid(long long work, int per, int cap) {
  long long b = (work + per - 1) / per;
  if (b < 1) b = 1;
  if (b > cap) b = cap;
  return (int)b;
}

extern "C" void kernel_launch(void* const* d_in, const int* in_sizes, int n_in,
                              void* d_out, int out_size, void* d_ws, size_t ws_size,
                              hipStream_t stream) {
  const float* emb = (const float*)d_in[0];
  const int* ei = (const int*)d_in[1];  // [2, E] int
  // d_in[2] (edge_attrs) is unused by the reference.

  const int n64 = in_sizes[0];  // N_NODES * 64
  const int n = n64 / 64;
  const int E = in_sizes[1] / 2;
  const int* frm = ei;
  const int* to = ei + E;

  // Workspace layout (256B aligned): deg/dinv | norm | xA | xB
  char* base = (char*)d_ws;
  size_t off = 0;
  float* deg = (float*)(base + off);
  off = (off + (size_t)n * 4 + 255) & ~(size_t)255;
  float* nrm = (float*)(base + off);
  off = (off + (size_t)E * 4 + 255) & ~(size_t)255;
  float* xA = (float*)(base + off);
  off += (size_t)n64 * 4;  // n64*4 is 256B-multiple
  float* xB = (float*)(base + off);

  float* out_emb0 = (float*)d_out;  // first tuple element
  float* acc = out_emb0 + n64;      // second tuple element (accumulator)

  const int TB = 256;
  const int CAP = 16384;

  // 1) degree
  k_zero<<<ngrid(n, TB * 4, CAP), TB, 0, stream>>>(deg, n);
  k_degree<<<ngrid(E, TB, CAP), TB, 0, stream>>>(to, deg, E);
  // 2) dinv = deg>0 ? rsqrt(deg) : 0   (in place)
  k_rsqrt_inplace<<<ngrid(n, TB, CAP), TB, 0, stream>>>(deg, n);
  // 3) per-edge norm
  k_norm<<<ngrid(E, TB, CAP), TB, 0, stream>>>(frm, to, deg, nrm, E);
  // 4) d_out: emb0 copy + accumulator init
  k_init_out<<<ngrid(n64 / 4, TB, CAP), TB, 0, stream>>>((const float4*)emb,
                                                         (float4*)d_out, n64 / 4);

  // 5) three propagation layers; acc lives in d_out second half
  const float* xcur = emb;
  float* bufs[2] = {xA, xB};
  const int gz = ngrid(n64, TB * 4, CAP);
  const int ga = ngrid(n64 / 4, TB, CAP);
  const int gp = ngrid(E, PROP_TILE, CAP);
  for (int layer = 0; layer < 3; ++layer) {
    float* xn = bufs[layer & 1];
    k_zero<<<gz, TB, 0, stream>>>(xn, n64);
    k_prop<<<gp, TB, 0, stream>>>(xcur, xn, frm, to, nrm, E);
    if (layer < 2) {
      k_accum_scale<<<ga, TB, 0, stream>>>((float4*)acc, (const float4*)xn,
                                           n64 / 4, 1.0f);
    }
    xcur = xn;
  }

  // 6) final layer-mean on the matrix pipe: acc = (acc + x3) * 0.25
  const int rowTiles = n / 16;
  if (rowTiles > 0) {
    const int waves = rowTiles * 4;
    k_wmma_finish<<<ngrid(waves, TB / 32, CAP), TB, 0, stream>>>(xcur, acc,
                                                                 rowTiles);
  }
  const int rows16 = rowTiles * 16;
  if (rows16 < n) {  // tail rows (none for n multiple of 16)
    const int tail4 = (n - rows16) * 64 / 4;
    k_accum_scale<<<ngrid(tail4, TB, CAP), TB, 0, stream>>>(
        (float4*)(acc + (size_t)rows16 * 64), (const float4*)(xcur + (size_t)rows16 * 64),
        tail4, 0.25f);
  }
}